// PoolbyIndices_2482491097343
// MI455X (gfx1250) — compile-verified
//
#include <hip/hip_runtime.h>
#include <hip/hip_bf16.h>

// Problem constants (from the reference)
#define BB   4
#define NN   2048
#define MM   8192
#define DD   128
#define NSEG 1024

// ---------------------------------------------------------------------------
// Kernel 1: zero the workspace (sums: B*NSEG*D floats, counts: B*NSEG floats).
// Must run every call: the scatter kernel accumulates into it.
// ---------------------------------------------------------------------------
__global__ __launch_bounds__(256) void zero_ws_kernel(float4* __restrict__ ws, int n4) {
    int i = blockIdx.x * blockDim.x + threadIdx.x;
    const float4 z = make_float4(0.f, 0.f, 0.f, 0.f);
    for (; i < n4; i += gridDim.x * blockDim.x) ws[i] = z;
}

// ---------------------------------------------------------------------------
// Kernel 2: segment scatter-reduce.
// One wave32 per source row. The wave async-copies its 512B row into LDS
// (one GLOBAL_LOAD_ASYNC_TO_LDS_B128 across 32 lanes), waits ASYNCcnt, reads
// the row back as float4 via ds_load_b128, then issues device-scope
// global_atomic_add_f32 into the L2-resident accumulator. Lane 0 also bumps
// the per-(b,seg) count by 1.0f.
// ---------------------------------------------------------------------------
__global__ __launch_bounds__(256) void scatter_segsum_kernel(
        const int*   __restrict__ idx_src,   // [B*M]
        const float* __restrict__ src,       // [B*M*D]
        float*       __restrict__ sums,      // [B*NSEG*D]
        float*       __restrict__ cnt) {     // [B*NSEG]
    __shared__ float smem[8 * DD];           // 8 waves * 128 floats = 4 KB

    const int wid  = threadIdx.x >> 5;
    const int lane = threadIdx.x & 31;
    const int r    = blockIdx.x * 8 + wid;   // global source row in [0, B*M)
    const int b    = r / MM;
    const int seg  = idx_src[r];             // wave-uniform

    // --- async global -> LDS copy of this wave's row (16 B per lane) ---
    const unsigned voff = (unsigned)r * (DD * 4u) + (unsigned)lane * 16u;
    const unsigned lds  = (unsigned)(size_t)(&smem[wid * DD + lane * 4]);
    asm volatile("global_load_async_to_lds_b128 %0, %1, %2"
                 :
                 : "v"(lds), "v"(voff), "s"(src)
                 : "memory");
    asm volatile("s_wait_asynccnt 0" ::: "memory");

    const float4 v = *(const float4*)(&smem[wid * DD + lane * 4]);

    // --- device-scope HW float atomics into the segment accumulator ---
    const unsigned soff = ((unsigned)(b * NSEG + seg) * DD + (unsigned)lane * 4u) * 4u;
    asm volatile("global_atomic_add_f32 %0, %1, %2 scope:SCOPE_DEV"
                 :: "v"(soff), "v"(v.x), "s"(sums) : "memory");
    asm volatile("global_atomic_add_f32 %0, %1, %2 offset:4 scope:SCOPE_DEV"
                 :: "v"(soff), "v"(v.y), "s"(sums) : "memory");
    asm volatile("global_atomic_add_f32 %0, %1, %2 offset:8 scope:SCOPE_DEV"
                 :: "v"(soff), "v"(v.z), "s"(sums) : "memory");
    asm volatile("global_atomic_add_f32 %0, %1, %2 offset:12 scope:SCOPE_DEV"
                 :: "v"(soff), "v"(v.w), "s"(sums) : "memory");

    if (lane == 0) {
        const unsigned coff = (unsigned)(b * NSEG + seg) * 4u;
        const float one = 1.0f;
        asm volatile("global_atomic_add_f32 %0, %1, %2 scope:SCOPE_DEV"
                     :: "v"(coff), "v"(one), "s"(cnt) : "memory");
    }
    // No-return atomics track STOREcnt; S_ENDPGM performs an implicit wait-idle.
}

// ---------------------------------------------------------------------------
// Kernel 3: gather + normalize. One wave32 per target row.
// sums/cnt (2 MB) are L2-resident; gather hits L2 at full rate.
// ---------------------------------------------------------------------------
__global__ __launch_bounds__(256) void gather_kernel(
        const int*   __restrict__ idx_tgt,   // [B*N]
        const float* __restrict__ sums,      // [B*NSEG*D]
        const float* __restrict__ cnt,       // [B*NSEG]
        float*       __restrict__ out) {     // [B*N*D]
    const int wid  = threadIdx.x >> 5;
    const int lane = threadIdx.x & 31;
    const int t    = blockIdx.x * 8 + wid;   // global target row in [0, B*N)
    const int b    = t / NN;
    const int seg  = idx_tgt[t];

    const float wsum = cnt[b * NSEG + seg] + 1e-10f;
    const float inv  = 1.0f / wsum;

    const float4 s = *(const float4*)(sums + (size_t)(b * NSEG + seg) * DD + lane * 4);
    float4 o;
    o.x = s.x * inv; o.y = s.y * inv; o.z = s.z * inv; o.w = s.w * inv;
    *(float4*)(out + (size_t)t * DD + lane * 4) = o;
}

// ---------------------------------------------------------------------------
// Host launcher. Inputs (setup_inputs order):
//   d_in[0] = index_target  int32 [B,N,1]
//   d_in[1] = index_source  int32 [B,M,1]
//   d_in[2] = array_source  f32   [B,M,D]
// d_out = f32 [B,N,D]; d_ws holds sums (2 MB) + counts (16 KB).
// ---------------------------------------------------------------------------
extern "C" void kernel_launch(void* const* d_in, const int* in_sizes, int n_in,
                              void* d_out, int out_size, void* d_ws, size_t ws_size,
                              hipStream_t stream) {
    const int*   idx_tgt = (const int*)d_in[0];
    const int*   idx_src = (const int*)d_in[1];
    const float* src     = (const float*)d_in[2];
    float*       out     = (float*)d_out;

    float* sums = (float*)d_ws;
    float* cnt  = sums + (size_t)BB * NSEG * DD;

    const int ws_floats4 = (BB * NSEG * DD + BB * NSEG) / 4;  // 132096 float4s
    zero_ws_kernel<<<256, 256, 0, stream>>>((float4*)d_ws, ws_floats4);

    scatter_segsum_kernel<<<(BB * MM) / 8, 256, 0, stream>>>(idx_src, src, sums, cnt);

    gather_kernel<<<(BB * NN) / 8, 256, 0, stream>>>(idx_tgt, sums, cnt, out);
}